// CollisionLoss_16844861735210
// MI455X (gfx1250) — compile-verified
//
#include <hip/hip_runtime.h>
#include <math.h>

// ---------------------------------------------------------------------------
// CollisionLoss for MI455X (gfx1250, wave32)
// Bandwidth-bound: 128MB single-pass read of future_gt_corners (~5.5us @23.3TB/s)
// Pipeline: abox precompute -> NT-load streaming reduce -> WMMA f32 tree reduce
// ---------------------------------------------------------------------------

typedef float f4  __attribute__((ext_vector_type(4)));
typedef float v2f __attribute__((ext_vector_type(2)));
typedef float v8f __attribute__((ext_vector_type(8)));

#define T_STEPS   256
#define N_AGENTS  16384
#define NITEMS    (T_STEPS * N_AGENTS)   // 4,194,304 (t,n) pairs
#define BLOCKS    2048
#define THREADS   256                    // 8 wave32s per block
#define NPART     BLOCKS                 // one partial per block

#define BOX_W 2.35f                      // 1.85 + 0.5
#define BOX_H 4.584f                     // 4.084 + 0.5

// --- Kernel 1: per-timestep SDC AABB (rotated base box -> axis-aligned box) ---
// max over the 4 rotated corners of +/-(c*hw) +/- (s*hh) is exactly |c|hw+|s|hh
// (bit-identical to the explicit 4-corner max since fp add is sign-symmetric).
__global__ void CollisionLoss_abox_kernel(const float* __restrict__ traj,
                                          const float* __restrict__ gt,
                                          const int*   __restrict__ mask,
                                          f4* __restrict__ abox) {
    int t = threadIdx.x;                 // one block of 256 threads
    float x   = traj[2 * t];
    float y   = traj[2 * t + 1];
    float yaw = gt[3 * t + 2];
    float c = cosf(yaw);
    float s = sinf(yaw);
    const float hw = BOX_W * 0.5f, hh = BOX_H * 0.5f;
    float dx = fabsf(c) * hw + fabsf(s) * hh;   // extent of rotated box in x
    float dy = fabsf(s) * hw + fabsf(c) * hh;   // extent of rotated box in y
    f4 box;
    if (mask[t] != 0) {
        box.x = x - dx;  box.y = y - dy;        // a_min
        box.z = x + dx;  box.w = y + dy;        // a_max
    } else {
        // inverted box -> hi-lo always negative -> clamped to 0 contribution
        box.x = 1e30f;  box.y = 1e30f;
        box.z = -1e30f; box.w = -1e30f;
    }
    abox[t] = box;
}

// --- Kernel 2: streaming AABB-intersection reduce over 128MB of corners ------
__global__ void CollisionLoss_inter_kernel(const f4* __restrict__ fc,
                                           const f4* __restrict__ abox,
                                           float* __restrict__ part) {
    int tid    = blockIdx.x * blockDim.x + threadIdx.x;
    int stride = gridDim.x * blockDim.x;
    float acc = 0.0f;
    for (int i = tid; i < NITEMS; i += stride) {
        // 8 contiguous floats per (t,n): [x0 y0 x1 y1][x2 y2 x3 y3]
        // Non-temporal: single-pass data, keep it out of the caches' hot set.
        f4 v0 = __builtin_nontemporal_load(&fc[2 * i]);
        f4 v1 = __builtin_nontemporal_load(&fc[2 * i + 1]);
        f4 a  = abox[i >> 14];           // t = i / 16384; uniform per wave -> WGP$ hit

        float bmaxx = fmaxf(fmaxf(v0.x, v0.z), fmaxf(v1.x, v1.z));
        float bminx = fminf(fminf(v0.x, v0.z), fminf(v1.x, v1.z));
        float bmaxy = fmaxf(fmaxf(v0.y, v0.w), fmaxf(v1.y, v1.w));
        float bminy = fminf(fminf(v0.y, v0.w), fminf(v1.y, v1.w));

        float ix = fmaxf(fminf(a.z, bmaxx) - fmaxf(a.x, bminx), 0.0f);
        float iy = fmaxf(fminf(a.w, bmaxy) - fmaxf(a.y, bminy), 0.0f);
        acc = fmaf(ix, iy, acc);
    }

    // wave32 tree reduce
    for (int off = 16; off > 0; off >>= 1)
        acc += __shfl_xor(acc, off);

    __shared__ float smem[THREADS / 32];
    int wid = threadIdx.x >> 5, lid = threadIdx.x & 31;
    if (lid == 0) smem[wid] = acc;
    __syncthreads();
    if (threadIdx.x == 0) {
        float sum = 0.0f;
        #pragma unroll
        for (int w = 0; w < THREADS / 32; ++w) sum += smem[w];
        part[blockIdx.x] = sum;          // deterministic: no atomics
    }
}

// --- Kernel 3: exact f32 WMMA tree reduction of the 2048 block partials ------
// A = all-ones 16x4  =>  D[m,n] = sum_k B[k,n] + C[m,n]  (column sums, all rows
// equal). Chaining 32 WMMAs accumulates all 2048 partials into the columns of C.
// c[0] holds rows M=0 (lanes 0-15) and M=8 (lanes 16-31), which are identical,
// so a full 32-lane sum of c[0] is exactly 2x the grand total.
__global__ void CollisionLoss_wmma_reduce_kernel(const float* __restrict__ part,
                                                 float* __restrict__ out) {
    int lane = threadIdx.x;              // exactly one wave32, EXEC all ones
    v8f c = {0.f, 0.f, 0.f, 0.f, 0.f, 0.f, 0.f, 0.f};
    v2f ones = {1.0f, 1.0f};
    for (int j = 0; j < NPART / 64; ++j) {
        v2f b;
        b.x = part[j * 64 + lane];
        b.y = part[j * 64 + 32 + lane];  // each partial appears exactly once in B
        // (neg_a, A, neg_b, B, c_mod, C, reuse_a, reuse_b)
        c = __builtin_amdgcn_wmma_f32_16x16x4_f32(false, ones, false, b,
                                                  (short)0, c, false, false);
    }
    float v = c[0];
    for (int off = 16; off > 0; off >>= 1)
        v += __shfl_xor(v, off);
    if (lane == 0) out[0] = 0.5f * v;    // WEIGHT == 1.0
}

extern "C" void kernel_launch(void* const* d_in, const int* in_sizes, int n_in,
                              void* d_out, int out_size, void* d_ws, size_t ws_size,
                              hipStream_t stream) {
    (void)in_sizes; (void)n_in; (void)out_size; (void)ws_size;
    const float* traj = (const float*)d_in[0];   // (1,256,2) f32
    const float* gt   = (const float*)d_in[1];   // (1,256,3) f32
    const int*   mask = (const int*)d_in[2];     // (1,256) i32
    const f4*    fc   = (const f4*)d_in[3];      // (256,16384,4,2) f32
    float*       out  = (float*)d_out;

    f4*    abox = (f4*)d_ws;                             // 256 * 16B = 4KB
    float* part = (float*)((char*)d_ws + 4096);          // 2048 * 4B = 8KB

    CollisionLoss_abox_kernel<<<1, T_STEPS, 0, stream>>>(traj, gt, mask, abox);
    CollisionLoss_inter_kernel<<<BLOCKS, THREADS, 0, stream>>>(fc, abox, part);
    CollisionLoss_wmma_reduce_kernel<<<1, 32, 0, stream>>>(part, out);
}